// MarkovJumpParticleFilter_29463475650751
// MI455X (gfx1250) — compile-verified
//
#include <hip/hip_runtime.h>
#include <math.h>

typedef __attribute__((ext_vector_type(2))) float v2f;
typedef __attribute__((ext_vector_type(4))) float v4f;
typedef __attribute__((ext_vector_type(8))) float v8f;

constexpr int DIM   = 64;
constexpr int LDST  = 65;   // padded LDS row stride (banks = 64)

// In-place lower Cholesky of a 64x64 LDS matrix (stride LDST).
// Leaves strictly-upper triangle untouched (callers must ignore/zero it).
// All 256 threads must call (contains __syncthreads).
__device__ __forceinline__ void chol_inplace(float* M, int tid) {
  for (int j = 0; j < DIM; ++j) {
    if (tid == 0) {
      float d = M[j * LDST + j];
      for (int k = 0; k < j; ++k) { float v = M[j * LDST + k]; d -= v * v; }
      M[j * LDST + j] = sqrtf(d);
    }
    __syncthreads();
    float inv = 1.0f / M[j * LDST + j];
    int i = j + 1 + tid;
    if (i < DIM) {
      float s = M[i * LDST + j];
      for (int k = 0; k < j; ++k) s -= M[i * LDST + k] * M[j * LDST + k];
      M[i * LDST + j] = s * inv;
    }
    __syncthreads();
  }
}

// One block per cluster. Produces:
//   wsCov[k]  = P - Kt^T P            (row-major [i][j])
//   wsGT[k]   = transpose(L - Kt^T L) (stored [i][j] = G[j][i] for coalesced matvec)
//   wsW[k]    = mu + Kt^T (obs - mu)
__global__ void __launch_bounds__(256)
cluster_precompute(const float* __restrict__ nodesMean,
                   const float* __restrict__ nodesCov,
                   const float* __restrict__ Rmat,
                   const float* __restrict__ obs,
                   float* __restrict__ wsCov,
                   float* __restrict__ wsGT,
                   float* __restrict__ wsW) {
  __shared__ float sP[DIM * LDST];   // P, later chol(P)=L
  __shared__ float sS[DIM * LDST];   // P+R, then chol -> Lc
  __shared__ float sKt[DIM * LDST];  // Kt = S^{-1} P
  __shared__ float sVec[DIM];        // obs - mu

  const int tid = threadIdx.x;
  const int k   = blockIdx.x;
  const float* Pg = nodesCov + (size_t)k * (DIM * DIM);

  for (int idx = tid; idx < DIM * DIM; idx += 256) {
    int r = idx >> 6, c = idx & 63;
    float p = Pg[idx];
    sP[r * LDST + c] = p;
    sS[r * LDST + c] = p + Rmat[idx];
  }
  if (tid < DIM) sVec[tid] = obs[tid] - nodesMean[k * DIM + tid];
  __syncthreads();

  chol_inplace(sS, tid);  // sS lower = Lc

  // Column-parallel triangular solves: S Kt = P  ->  Kt = S^{-1} P
  if (tid < DIM) {
    const int j = tid;
    for (int i = 0; i < DIM; ++i) {                 // forward: Lc y = P(:,j)
      float s = sP[i * LDST + j];
      for (int kk = 0; kk < i; ++kk) s -= sS[i * LDST + kk] * sKt[kk * LDST + j];
      sKt[i * LDST + j] = s / sS[i * LDST + i];
    }
    for (int i = DIM - 1; i >= 0; --i) {            // backward: Lc^T x = y
      float s = sKt[i * LDST + j];
      for (int kk = i + 1; kk < DIM; ++kk) s -= sS[kk * LDST + i] * sKt[kk * LDST + j];
      sKt[i * LDST + j] = s / sS[i * LDST + i];
    }
    float acc = nodesMean[k * DIM + j];             // w = mu + Kt^T (obs-mu)
    for (int i = 0; i < DIM; ++i) acc += sKt[i * LDST + j] * sVec[i];
    wsW[k * DIM + j] = acc;
  }
  __syncthreads();

  // Fragment geometry (16x16x4 f32 WMMA, wave32)
  const int wave  = tid >> 5;
  const int lane  = tid & 31;
  const int m16   = lane & 15;
  const int khalf = (lane >> 4) << 1;  // K sub-offset: 0 or 2
  const int roff  = (lane >> 4) << 3;  // C/D row offset: 0 or 8

  // ---- Matmul 1: updCov = P - Kt^T P  (A = -Kt^T, B = P, C = P) ----
  for (int t = 0; t < 2; ++t) {
    const int tt = wave * 2 + t;                 // 16 tiles over 8 waves
    const int ti = tt >> 2, tj = tt & 3;
    const int mA = ti * 16 + m16;                // A row (output row block)
    const int nB = tj * 16 + m16;                // B/C/D column
    v8f c;
#pragma unroll
    for (int r = 0; r < 8; ++r)
      c[r] = sP[(ti * 16 + r + roff) * LDST + nB];
#pragma unroll
    for (int k0 = 0; k0 < DIM; k0 += 4) {
      v2f a, b;
      a.x = -sKt[(k0 + khalf) * LDST + mA];
      a.y = -sKt[(k0 + khalf + 1) * LDST + mA];
      b.x =  sP[(k0 + khalf) * LDST + nB];
      b.y =  sP[(k0 + khalf + 1) * LDST + nB];
      c = __builtin_amdgcn_wmma_f32_16x16x4_f32(false, a, false, b,
                                                (short)0, c, false, false);
    }
    float* dst = wsCov + (size_t)k * (DIM * DIM);
#pragma unroll
    for (int r = 0; r < 8; ++r)
      dst[(ti * 16 + r + roff) * DIM + nB] = c[r];
  }
  __syncthreads();  // all reads of sP done before overwriting it

  // ---- chol(P) -> L in place; zero strictly-upper so L is dense-correct ----
  chol_inplace(sP, tid);
  for (int idx = tid; idx < DIM * DIM; idx += 256) {
    int r = idx >> 6, c2 = idx & 63;
    if (c2 > r) sP[r * LDST + c2] = 0.0f;
  }
  __syncthreads();

  // ---- Matmul 2: G = L - Kt^T L  (A = -Kt^T, B = L, C = L); store G^T ----
  for (int t = 0; t < 2; ++t) {
    const int tt = wave * 2 + t;
    const int ti = tt >> 2, tj = tt & 3;
    const int mA = ti * 16 + m16;
    const int nB = tj * 16 + m16;
    v8f c;
#pragma unroll
    for (int r = 0; r < 8; ++r)
      c[r] = sP[(ti * 16 + r + roff) * LDST + nB];
#pragma unroll
    for (int k0 = 0; k0 < DIM; k0 += 4) {
      v2f a, b;
      a.x = -sKt[(k0 + khalf) * LDST + mA];
      a.y = -sKt[(k0 + khalf + 1) * LDST + mA];
      b.x =  sP[(k0 + khalf) * LDST + nB];
      b.y =  sP[(k0 + khalf + 1) * LDST + nB];
      c = __builtin_amdgcn_wmma_f32_16x16x4_f32(false, a, false, b,
                                                (short)0, c, false, false);
    }
    float* dstG = wsGT + (size_t)k * (DIM * DIM);
#pragma unroll
    for (int r = 0; r < 8; ++r)  // GT[n][m] = G[m][n]
      dstG[nB * DIM + (ti * 16 + r + roff)] = c[r];
  }
}

// One block per particle: stream the 16KB cov broadcast (NT stores) and
// one 64x64 matvec for the mean (coalesced reads of L2-resident G^T).
__global__ void __launch_bounds__(256)
particle_kernel(const float* __restrict__ z,
                const int* __restrict__ assign,
                const float* __restrict__ wsCov,
                const float* __restrict__ wsGT,
                const float* __restrict__ wsW,
                float* __restrict__ outMeans,
                float* __restrict__ outCovs) {
  __shared__ float zs[DIM];
  const int p   = blockIdx.x;
  const int tid = threadIdx.x;
  const int c   = assign[p];

  if (tid < DIM) zs[tid] = z[(size_t)p * DIM + tid];
  __syncthreads();

  // cov' gather-broadcast: 1024 16B vectors per particle, non-temporal writes
  const v4f* src = (const v4f*)(wsCov + (size_t)c * (DIM * DIM));
  v4f* dst = (v4f*)(outCovs + (size_t)p * (DIM * DIM));
#pragma unroll
  for (int i = 0; i < 4; ++i) {
    v4f v = src[tid + i * 256];
    __builtin_nontemporal_store(v, dst + tid + i * 256);
  }

  // mean' = w_c + G_c z  (thread j owns output j; GT reads coalesced)
  if (tid < DIM) {
    float acc = wsW[c * DIM + tid];
    const float* GT = wsGT + (size_t)c * (DIM * DIM);
#pragma unroll 8
    for (int i = 0; i < DIM; ++i) acc += GT[i * DIM + tid] * zs[i];
    outMeans[(size_t)p * DIM + tid] = acc;
  }
}

extern "C" void kernel_launch(void* const* d_in, const int* in_sizes, int n_in,
                              void* d_out, int out_size, void* d_ws, size_t ws_size,
                              hipStream_t stream) {
  const float* nodesMean = (const float*)d_in[0];   // [K,64]
  const float* nodesCov  = (const float*)d_in[1];   // [K,64,64]
  const float* obsCov    = (const float*)d_in[2];   // [64,64]
  const float* z         = (const float*)d_in[3];   // [N,64]
  const float* obsState  = (const float*)d_in[4];   // [64]
  const int*   assign    = (const int*)d_in[5];     // [N]

  const int K = in_sizes[0] / DIM;
  const int N = in_sizes[5];

  float* wsCov = (float*)d_ws;                       // K*4096 floats
  float* wsGT  = wsCov + (size_t)K * DIM * DIM;      // K*4096 floats
  float* wsW   = wsGT + (size_t)K * DIM * DIM;       // K*64 floats

  float* outMeans = (float*)d_out;                   // N*64
  float* outCovs  = outMeans + (size_t)N * DIM;      // N*4096

  hipLaunchKernelGGL(cluster_precompute, dim3(K), dim3(256), 0, stream,
                     nodesMean, nodesCov, obsCov, obsState, wsCov, wsGT, wsW);
  hipLaunchKernelGGL(particle_kernel, dim3(N), dim3(256), 0, stream,
                     z, assign, wsCov, wsGT, wsW, outMeans, outCovs);
}